// HeteroDotProductPredictor_63187558858871
// MI455X (gfx1250) — compile-verified
//
#include <hip/hip_runtime.h>

// CDNA5 / gfx1250: wave32, WMMA 16x16x4 f32.
typedef __attribute__((ext_vector_type(2))) float v2f;
typedef __attribute__((ext_vector_type(8))) float v8f;

#define D_FEAT 128
#define EDGES_PER_WAVE 16
#define WAVES_PER_BLOCK 8

// One wave computes 16 edge dot-products as the diagonal of a 16x16 WMMA
// result, accumulating D = A(16x4) x B(4x16) + C over 32 K-chunks.
//
// Per-lane operand layout for V_WMMA_F32_16X16X4_F32 (wave32):
//   A: lane L<16 holds A[L][{0,1}] in vgpr{0,1}; lane L>=16 holds A[L-16][{2,3}]
//   B: lane L<16 holds B[{0,1}][L];              lane L>=16 holds B[{2,3}][L-16]
// We exploit K-permutation invariance of the dot product: lanes 0-15 stream
// feature elements 0..63 and lanes 16-31 stream elements 64..127, so every
// lane does contiguous 8-byte loads from its row.
__global__ __launch_bounds__(WAVES_PER_BLOCK * 32)
void edge_dot_wmma_f32(const float* __restrict__ h,
                       const int* __restrict__ src,
                       const int* __restrict__ dst,
                       float* __restrict__ out,
                       int n_edges)
{
    const int lane = threadIdx.x & 31;
    const int wave = threadIdx.x >> 5;
    const int tile = blockIdx.x * WAVES_PER_BLOCK + wave;  // 16 edges / tile
    const int base = tile * EDGES_PER_WAVE;
    if (base >= n_edges) return;  // uniform per wave -> whole wave exits

    // Edge slot this lane serves (two lanes per edge: one per feature half).
    const int slot = lane & 15;
    int e = base + slot;
    if (e >= n_edges) e = n_edges - 1;  // clamp: keep EXEC all-1s for WMMA

    const int si = src[e];
    const int di = dst[e];

    const int half = lane >> 4;  // 0 -> elements 0..63, 1 -> elements 64..127
    const float* __restrict__ rowA = h + (size_t)si * D_FEAT + half * 64;
    const float* __restrict__ rowB = h + (size_t)di * D_FEAT + half * 64;

    v8f acc = {};
#pragma unroll
    for (int c = 0; c < 32; ++c) {
        v2f a = *(const v2f*)(rowA + 2 * c);   // global_load_b64
        v2f b = *(const v2f*)(rowB + 2 * c);   // global_load_b64
        // 8 args: (neg_a, A, neg_b, B, c_mod, C, reuse_a, reuse_b)
        acc = __builtin_amdgcn_wmma_f32_16x16x4_f32(
            false, a, false, b, (short)0, acc, false, false);
    }

    // D (16x16 f32) layout: vgpr v, lanes 0-15 -> M=v, N=lane;
    //                       lanes 16-31 -> M=v+8, N=lane-16.
    // Diagonal M==N lives in: lane m (vgpr m) for m=0..7,
    //                         lane m+24-8.. i.e. lane = m+16 with vgpr m-8 for m=8..15.
    int m = -1;
    int v = 0;
    if (lane < 8)        { m = lane;      v = lane;      }
    else if (lane >= 24) { m = lane - 16; v = lane - 24; }

    float val = acc[0];
    val = (v == 1) ? acc[1] : val;
    val = (v == 2) ? acc[2] : val;
    val = (v == 3) ? acc[3] : val;
    val = (v == 4) ? acc[4] : val;
    val = (v == 5) ? acc[5] : val;
    val = (v == 6) ? acc[6] : val;
    val = (v == 7) ? acc[7] : val;

    const int eo = base + m;
    if (m >= 0 && eo < n_edges) out[eo] = val;
}

extern "C" void kernel_launch(void* const* d_in, const int* in_sizes, int n_in,
                              void* d_out, int out_size, void* d_ws, size_t ws_size,
                              hipStream_t stream) {
    const float* h   = (const float*)d_in[0];
    const int*   src = (const int*)d_in[1];
    const int*   dst = (const int*)d_in[2];
    float*       out = (float*)d_out;

    const int n_edges = in_sizes[1];
    const int tiles   = (n_edges + EDGES_PER_WAVE - 1) / EDGES_PER_WAVE;
    const int blocks  = (tiles + WAVES_PER_BLOCK - 1) / WAVES_PER_BLOCK;

    edge_dot_wmma_f32<<<blocks, WAVES_PER_BLOCK * 32, 0, stream>>>(
        h, src, dst, out, n_edges);
}